// BiGlobal_MPCMFuse_36687610642742
// MI455X (gfx1250) — compile-verified
//
#include <hip/hip_runtime.h>
#include <hip/hip_bf16.h>
#include <math.h>

#define H 384
#define W 384
#define BATCH 4
#define C 64
#define IC 32
#define TILE_ROWS 32
#define TILES (H / TILE_ROWS)              /* 12 */
#define HALO 17
#define SLAB_ROWS (TILE_ROWS + 2 * HALO)   /* 66 */
#define EPS 1e-5f

typedef __attribute__((ext_vector_type(2))) float v2f;
typedef __attribute__((ext_vector_type(8))) float v8f;
typedef __attribute__((address_space(3))) float lds_f32;

/* workspace layout (in floats):
   [0 .. B*C*TILES*4)          per-block partials {sum13, sq13, sum17, sq17}
   [WS_STATS .. +4*C)          a13[64] b13[64] a17[64] b17[64]  (BN affine)
   [WS_WEI .. +2*B*C)          td_wei[256], bu_wei[256]                        */
#define WS_STATS (BATCH * C * TILES * 4)
#define WS_WEI (WS_STATS + 4 * C)

/* Stage halo slab global -> LDS with CDNA5 async-to-LDS (no VGPR staging).
   Per lane: VDST = LDS byte offset, VADDR = 64-bit global address.           */
__device__ __forceinline__ void load_slab_async(const float* __restrict__ plane,
                                                int r0, float* slab) {
  const int nvec = SLAB_ROWS * (W / 4); /* 66*96 = 6336 b128 transfers */
  for (int q = threadIdx.x; q < nvec; q += blockDim.x) {
    int row = q / (W / 4);
    int c4 = q - row * (W / 4);
    int gy = r0 - HALO + row;
    if (gy < 0) gy += H;
    if (gy >= H) gy -= H;
    const float* gptr = plane + gy * W + c4 * 4;
    unsigned lds_off = (unsigned)(unsigned long long)(lds_f32*)(slab + row * W + c4 * 4);
    asm volatile("global_load_async_to_lds_b128 %0, %1, off"
                 :: "v"(lds_off), "v"(gptr) : "memory");
  }
  asm volatile("s_wait_asynccnt 0" ::: "memory");
}

/* pcm at (ly, x) in slab coordinates; roll(-s,-s) == cen[y+s, x+s] */
__device__ __forceinline__ float pcm_at(const float* slab, int ly, int x, int s) {
  const float c = slab[ly * W + x];
  int xm = x - s; if (xm < 0) xm += W;
  int xp = x + s; if (xp >= W) xp -= W;
  const float* up = slab + (ly + s) * W; /* y+s */
  const float* dn = slab + (ly - s) * W; /* y-s */
  const float* md = slab + ly * W;
  float s1 = (up[xp] - c) * (dn[xm] - c);
  float s2 = (up[x]  - c) * (dn[x]  - c);
  float s3 = (up[xm] - c) * (dn[xp] - c);
  float s4 = (md[xm] - c) * (md[xp] - c);
  return fminf(fminf(s1, s2), fminf(s3, s4));
}

/* ---------------- Pass 1: stencil + partial sums for BN stats ---------------- */
__global__ void pcm_stats_kernel(const float* __restrict__ cen, float* __restrict__ ws) {
  extern __shared__ float slab[];
  const int tile = blockIdx.x;
  const int plane = blockIdx.y;
  const int r0 = tile * TILE_ROWS;
  const float* plane_ptr = cen + (size_t)plane * H * W;

  load_slab_async(plane_ptr, r0, slab);
  __syncthreads();

  float s13 = 0.f, q13 = 0.f, s17 = 0.f, q17 = 0.f;
  for (int e = threadIdx.x; e < TILE_ROWS * W; e += blockDim.x) {
    int ly = e / W + HALO;
    int x = e - (ly - HALO) * W;
    float p13 = pcm_at(slab, ly, x, 13);
    float p17 = pcm_at(slab, ly, x, 17);
    s13 += p13; q13 += p13 * p13;
    s17 += p17; q17 += p17 * p17;
  }
  __syncthreads(); /* slab reuse as reduction scratch */

  float* red = slab;
  red[threadIdx.x] = s13;
  red[256 + threadIdx.x] = q13;
  red[512 + threadIdx.x] = s17;
  red[768 + threadIdx.x] = q17;
  __syncthreads();
  for (int off = 128; off > 0; off >>= 1) {
    if (threadIdx.x < off) {
      red[threadIdx.x] += red[threadIdx.x + off];
      red[256 + threadIdx.x] += red[256 + threadIdx.x + off];
      red[512 + threadIdx.x] += red[512 + threadIdx.x + off];
      red[768 + threadIdx.x] += red[768 + threadIdx.x + off];
    }
    __syncthreads();
  }
  if (threadIdx.x == 0) {
    float* p = ws + (size_t)(plane * TILES + tile) * 4;
    p[0] = red[0]; p[1] = red[256]; p[2] = red[512]; p[3] = red[768];
  }
}

/* SE branch: pooled P (16x64 in LDS, rows 4..15 zero) -> conv1(WMMA) -> BN+relu
   -> conv2(WMMA) -> BN+sigmoid -> wei[4*64]. Conv biases cancel under BN.
   __forceinline__ so P/Y1/Y2 stay addrspace(3) (ds_load) and w1/w2 stay
   addrspace(1) (global_load) — no generic/flat fallback.                      */
__device__ __forceinline__ void se_branch(const float* P, const float* __restrict__ w1,
                                          const float* __restrict__ g1,
                                          const float* __restrict__ be1,
                                          const float* __restrict__ w2,
                                          const float* __restrict__ g2,
                                          const float* __restrict__ be2,
                                          float* __restrict__ wei_out,
                                          float* Y1, float* Y2) {
  const int t = threadIdx.x;
  __syncthreads();

  /* conv1: D[16x32] = P[16x64] x w1^T[64x32], wave 0 only, f32 WMMA K=4 steps */
  if ((t >> 5) == 0) {
    const int lane = t;
    const int m = lane & 15;
    const int hi = lane >> 4;
    for (int ntile = 0; ntile < 2; ++ntile) {
      v8f acc = {};
      const int n = ntile * 16 + m;
      for (int kk = 0; kk < 16; ++kk) {
        const int k0 = kk * 4 + 2 * hi;
        v2f a, b;
        a.x = P[m * C + k0];      a.y = P[m * C + k0 + 1];
        b.x = w1[n * C + k0];     b.y = w1[n * C + k0 + 1];
        acc = __builtin_amdgcn_wmma_f32_16x16x4_f32(false, a, false, b,
                                                    (short)0, acc, false, false);
      }
      for (int v = 0; v < 8; ++v)
        Y1[(v + 8 * hi) * IC + ntile * 16 + m] = acc[v];
    }
  }
  __syncthreads();

  /* batch-only BN (population var over 4) + ReLU; keep pad rows at zero */
  float y1new = 0.f;
  if (t < BATCH * IC) {
    const int ic = t & (IC - 1);
    float x0 = Y1[0 * IC + ic], x1 = Y1[1 * IC + ic];
    float x2 = Y1[2 * IC + ic], x3 = Y1[3 * IC + ic];
    float mme = 0.25f * (x0 + x1 + x2 + x3);
    float var = 0.25f * (x0 * x0 + x1 * x1 + x2 * x2 + x3 * x3) - mme * mme;
    float aa = g1[ic] * rsqrtf(var + EPS);
    y1new = fmaxf((Y1[t] - mme) * aa + be1[ic], 0.f);
  }
  __syncthreads();
  Y1[t] = (t < BATCH * IC) ? y1new : 0.f; /* rows 0..7 */
  Y1[t + 256] = 0.f;                      /* rows 8..15 */
  __syncthreads();

  /* conv2: D[16x64] = Y1[16x32] x w2^T[32x64] */
  if ((t >> 5) == 0) {
    const int lane = t;
    const int m = lane & 15;
    const int hi = lane >> 4;
    for (int ntile = 0; ntile < 4; ++ntile) {
      v8f acc = {};
      const int n = ntile * 16 + m;
      for (int kk = 0; kk < 8; ++kk) {
        const int k0 = kk * 4 + 2 * hi;
        v2f a, b;
        a.x = Y1[m * IC + k0];    a.y = Y1[m * IC + k0 + 1];
        b.x = w2[n * IC + k0];    b.y = w2[n * IC + k0 + 1];
        acc = __builtin_amdgcn_wmma_f32_16x16x4_f32(false, a, false, b,
                                                    (short)0, acc, false, false);
      }
      for (int v = 0; v < 8; ++v)
        Y2[(v + 8 * hi) * C + ntile * 16 + m] = acc[v];
    }
  }
  __syncthreads();

  /* batch-only BN + sigmoid; t = b*64 + c covers all 256 outputs */
  {
    const int c = t & (C - 1);
    float x0 = Y2[0 * C + c], x1 = Y2[1 * C + c];
    float x2 = Y2[2 * C + c], x3 = Y2[3 * C + c];
    float mme = 0.25f * (x0 + x1 + x2 + x3);
    float var = 0.25f * (x0 * x0 + x1 * x1 + x2 * x2 + x3 * x3) - mme * mme;
    float aa = g2[c] * rsqrtf(var + EPS);
    float z = (Y2[t] - mme) * aa + be2[c];
    wei_out[t] = 1.f / (1.f + expf(-z));
  }
  __syncthreads();
}

/* ---------------- Pass 2 (tiny): BN stats + SE attention via WMMA ------------ */
__global__ void se_stats_kernel(float* __restrict__ ws,
                                const float* __restrict__ bn1_g, const float* __restrict__ bn1_b,
                                const float* __restrict__ bn2_g, const float* __restrict__ bn2_b,
                                const float* __restrict__ td_w1, const float* __restrict__ td_g1,
                                const float* __restrict__ td_be1, const float* __restrict__ td_w2,
                                const float* __restrict__ td_g2, const float* __restrict__ td_be2,
                                const float* __restrict__ bu_w1, const float* __restrict__ bu_g1,
                                const float* __restrict__ bu_be1, const float* __restrict__ bu_w2,
                                const float* __restrict__ bu_g2, const float* __restrict__ bu_be2) {
  __shared__ float sum13[BATCH * C], sqa13[BATCH * C];
  __shared__ float sum17[BATCH * C], sqa17[BATCH * C];
  __shared__ float a13[C], b13[C], a17[C], b17[C];
  __shared__ float P13[16 * C], P17[16 * C];
  __shared__ float Y1[16 * IC], Y2[16 * C];

  const int t = threadIdx.x;

  /* reduce tile partials -> per-(b,c) plane sums */
  float s13 = 0.f, q13 = 0.f, s17 = 0.f, q17 = 0.f;
  for (int tile = 0; tile < TILES; ++tile) {
    const float* p = ws + (size_t)(t * TILES + tile) * 4;
    s13 += p[0]; q13 += p[1]; s17 += p[2]; q17 += p[3];
  }
  sum13[t] = s13; sqa13[t] = q13; sum17[t] = s17; sqa17[t] = q17;
  __syncthreads();

  /* per-channel stats over (B,H,W) -> BN affine a*x+b */
  if (t < C) {
    const float Nn = (float)BATCH * (float)H * (float)W;
    float ts = 0.f, tq = 0.f;
    for (int b = 0; b < BATCH; ++b) { ts += sum13[b * C + t]; tq += sqa13[b * C + t]; }
    float m = ts / Nn, v = tq / Nn - m * m;
    float a = bn1_g[t] * rsqrtf(v + EPS);
    a13[t] = a; b13[t] = bn1_b[t] - m * a;
    ts = 0.f; tq = 0.f;
    for (int b = 0; b < BATCH; ++b) { ts += sum17[b * C + t]; tq += sqa17[b * C + t]; }
    m = ts / Nn; v = tq / Nn - m * m;
    a = bn2_g[t] * rsqrtf(v + EPS);
    a17[t] = a; b17[t] = bn2_b[t] - m * a;
  }
  __syncthreads();

  /* pooled, BN'd SE inputs as 16x64 matrices (rows 4..15 zero) */
  const float invHW = 1.f / ((float)H * (float)W);
  for (int i = t; i < 16 * C; i += 256) {
    float v13 = 0.f, v17 = 0.f;
    if (i < BATCH * C) {
      int c = i & (C - 1);
      v13 = a13[c] * (sum13[i] * invHW) + b13[c];
      v17 = a17[c] * (sum17[i] * invHW) + b17[c];
    }
    P13[i] = v13; P17[i] = v17;
  }
  if (t < C) { /* persist BN affine for the fuse pass */
    float* st = ws + WS_STATS;
    st[t] = a13[t]; st[C + t] = b13[t]; st[2 * C + t] = a17[t]; st[3 * C + t] = b17[t];
  }
  __syncthreads();

  /* topdown from pcm17, bottomup from pcm13 */
  se_branch(P17, td_w1, td_g1, td_be1, td_w2, td_g2, td_be2, ws + WS_WEI, Y1, Y2);
  se_branch(P13, bu_w1, bu_g1, bu_be1, bu_w2, bu_g2, bu_be2, ws + WS_WEI + BATCH * C, Y1, Y2);
}

/* ---------------- Pass 3: recompute stencil, apply BN + SE weights ----------- */
__global__ void fuse_kernel(const float* __restrict__ cen, const float* __restrict__ ws,
                            float* __restrict__ out) {
  extern __shared__ float slab[];
  const int tile = blockIdx.x;
  const int plane = blockIdx.y;
  const int c = plane & (C - 1);
  const int r0 = tile * TILE_ROWS;

  const float* st = ws + WS_STATS;
  const float A13 = st[c], B13 = st[C + c], A17 = st[2 * C + c], B17 = st[3 * C + c];
  const float tdw = ws[WS_WEI + plane];
  const float buw = ws[WS_WEI + BATCH * C + plane];

  const float* plane_ptr = cen + (size_t)plane * H * W;
  load_slab_async(plane_ptr, r0, slab);
  __syncthreads();

  float* oplane = out + (size_t)plane * H * W + (size_t)r0 * W;
  for (int e = threadIdx.x; e < TILE_ROWS * W; e += blockDim.x) {
    int ly = e / W + HALO;
    int x = e - (ly - HALO) * W;
    float p13 = A13 * pcm_at(slab, ly, x, 13) + B13;
    float p17 = A17 * pcm_at(slab, ly, x, 17) + B17;
    oplane[e] = tdw * p13 + buw * p17;
  }
}

extern "C" void kernel_launch(void* const* d_in, const int* in_sizes, int n_in,
                              void* d_out, int out_size, void* d_ws, size_t ws_size,
                              hipStream_t stream) {
  const float* cen    = (const float*)d_in[0];
  const float* bn1_g  = (const float*)d_in[1];
  const float* bn1_b  = (const float*)d_in[2];
  const float* bn2_g  = (const float*)d_in[3];
  const float* bn2_b  = (const float*)d_in[4];
  const float* td_w1  = (const float*)d_in[5];
  /* td_b1 = d_in[6] unused: conv bias cancels under training-mode BN */
  const float* td_g1  = (const float*)d_in[7];
  const float* td_be1 = (const float*)d_in[8];
  const float* td_w2  = (const float*)d_in[9];
  /* td_b2 = d_in[10] unused */
  const float* td_g2  = (const float*)d_in[11];
  const float* td_be2 = (const float*)d_in[12];
  const float* bu_w1  = (const float*)d_in[13];
  /* bu_b1 = d_in[14] unused */
  const float* bu_g1  = (const float*)d_in[15];
  const float* bu_be1 = (const float*)d_in[16];
  const float* bu_w2  = (const float*)d_in[17];
  /* bu_b2 = d_in[18] unused */
  const float* bu_g2  = (const float*)d_in[19];
  const float* bu_be2 = (const float*)d_in[20];

  float* ws = (float*)d_ws;
  float* out = (float*)d_out;

  dim3 blk(256);
  dim3 grd(TILES, BATCH * C);
  size_t slab_bytes = (size_t)SLAB_ROWS * W * sizeof(float); /* 101,376 B < 320 KB WGP LDS */

  pcm_stats_kernel<<<grd, blk, slab_bytes, stream>>>(cen, ws);
  se_stats_kernel<<<1, 256, 0, stream>>>(ws, bn1_g, bn1_b, bn2_g, bn2_b,
                                         td_w1, td_g1, td_be1, td_w2, td_g2, td_be2,
                                         bu_w1, bu_g1, bu_be1, bu_w2, bu_g2, bu_be2);
  fuse_kernel<<<grd, blk, slab_bytes, stream>>>(cen, ws, out);
}